// arbLoss_14886356648770
// MI455X (gfx1250) — compile-verified
//
#include <hip/hip_runtime.h>

typedef __attribute__((ext_vector_type(2))) float v2f;
typedef __attribute__((ext_vector_type(8))) float v8f;

#define NN 10000
#define DD 256
#define KK 5000
#define NTILES 625        // NN / 16
#define ASTRIDE 260       // 256 + 4 pad -> stride%64==4 -> conflict-free ds_load_b64
#define ROWS_PER_CS 250   // colsum rows per workgroup (40 WGs)

// ws layout (floats): [0]=dirichlet [1]=sumsq [2]=colsq [3]=recon [4..259]=colacc
#define WS_FLOATS 260

__global__ __launch_bounds__(512) void zero_ws_kernel(float* ws) {
    int t = threadIdx.x;
    if (t < WS_FLOATS) ws[t] = 0.0f;
}

// ---------------------------------------------------------------------------
// dirichlet = sum_ij L_ij * <xn_i, xn_j> = sum_ij L_ij * (<xn_i, x_j> - q_i),
// q_i = <xn_i, mean>.  Upper-triangle tiles only (L, Gram both symmetric):
// off-diagonal tiles weighted 2x.  WMMA f32 16x16x4 Gram accumulation over D.
// grid: (625 i-blocks) x (2 j-parity) x 256 threads (8 waves)
// ---------------------------------------------------------------------------
__global__ __launch_bounds__(256) void gram_dirichlet_kernel(
    const float* __restrict__ x, const float* __restrict__ L,
    const float* __restrict__ mean, float* __restrict__ ws)
{
    __shared__ float aT[16 * ASTRIDE];   // mean-subtracted i-block, padded rows
    __shared__ float ms[DD];
    __shared__ float qs[16];             // q[r] = <xn_{i0+r}, mean>

    const int tid = threadIdx.x;
    const int it  = blockIdx.x;
    const int par = blockIdx.y;          // parity of (jt - it)
    const int i0  = it * 16;

    if (tid < DD) ms[tid] = mean[tid];
    if (tid < 16) qs[tid] = 0.0f;
    for (int e = tid; e < 16 * DD; e += 256) {
        int rr = e >> 8, cc = e & 255;
        aT[rr * ASTRIDE + cc] = x[(size_t)(i0 + rr) * DD + cc] - mean[cc];
    }
    __syncthreads();

    // q[r] = <xn_r, mean>: 16 threads per row, 16-elem partials, LDS atomics
    {
        int rr = tid >> 4;
        int c0 = (tid & 15) * 16;
        float p = 0.f;
        #pragma unroll
        for (int c = 0; c < 16; ++c)
            p += aT[rr * ASTRIDE + c0 + c] * ms[c0 + c];
        atomicAdd(&qs[rr], p);
    }
    __syncthreads();

    const int wave = tid >> 5;
    const int lane = tid & 31;
    const int r    = lane & 15;    // M (A) / N (B) index within tile
    const int hi   = lane >> 4;    // lane group -> K offset
    const int koff = hi * 2;

    float qv[8];
    #pragma unroll
    for (int v = 0; v < 8; ++v) qv[v] = qs[v + 8 * hi];  // q at row M = v + 8*hi

    float part = 0.0f;

    for (int jt = it + par + 2 * wave; jt < NTILES; jt += 16) {
        const int j0 = jt * 16;
        v8f cacc = {0.f, 0.f, 0.f, 0.f, 0.f, 0.f, 0.f, 0.f};

        const float* ap = &aT[r * ASTRIDE + koff];            // LDS, bank-perfect
        const float* bp = x + (size_t)(j0 + r) * DD + koff;   // raw x, L2-resident

        #pragma unroll 8
        for (int k = 0; k < DD; k += 4) {
            v2f a = *(const v2f*)(ap + k);
            v2f b = *(const v2f*)(bp + k);
            // A: 16x4 f32 (xn i-block), B: 4x16 f32 (raw x j-block), C: f32 acc
            cacc = __builtin_amdgcn_wmma_f32_16x16x4_f32(
                false, a, false, b, (short)0, cacc, false, false);
        }

        // C layout: VGPR v -> M = v + 8*hi, lane%16 -> N.  L is streamed once:
        // non-temporal loads keep xn resident in L2.
        const float* lrow = L + (size_t)(i0 + 8 * hi) * NN + (size_t)(j0 + r);
        float tp = 0.f;
        #pragma unroll
        for (int v = 0; v < 8; ++v)
            tp += (cacc[v] - qv[v]) * __builtin_nontemporal_load(lrow + (size_t)v * NN);

        part += (jt == it) ? tp : 2.0f * tp;   // mirror tile via symmetry
    }

    for (int off = 16; off > 0; off >>= 1)
        part += __shfl_down(part, off, 32);
    if (lane == 0) atomicAdd(&ws[0], part);
}

// ---------------------------------------------------------------------------
// per-column raw sums + sum((x-mean)^2); 40 WGs x 256 threads (1 col/thread)
// ---------------------------------------------------------------------------
__global__ __launch_bounds__(256) void colsum_kernel(
    const float* __restrict__ x, const float* __restrict__ mean,
    float* __restrict__ ws)
{
    const int d  = threadIdx.x;
    const int r0 = blockIdx.x * ROWS_PER_CS;
    const float m = mean[d];
    float s = 0.f, sq = 0.f;
    for (int i = 0; i < ROWS_PER_CS; ++i) {
        float v  = x[(size_t)(r0 + i) * DD + d];
        float vn = v - m;
        s  += v;
        sq += vn * vn;
    }
    atomicAdd(&ws[4 + d], s);
    for (int off = 16; off > 0; off >>= 1)
        sq += __shfl_down(sq, off, 32);
    if ((threadIdx.x & 31) == 0) atomicAdd(&ws[1], sq);
}

__global__ __launch_bounds__(256) void colsq_kernel(
    const float* __restrict__ mean, float* __restrict__ ws)
{
    const int d = threadIdx.x;
    float col = ws[4 + d] - (float)NN * mean[d];
    float c2  = col * col;
    for (int off = 16; off > 0; off >>= 1)
        c2 += __shfl_down(c2, off, 32);
    if ((threadIdx.x & 31) == 0) atomicAdd(&ws[2], c2);
}

// ---------------------------------------------------------------------------
// recon = sum over known rows of (xn[row] - out_k_init)^2
// ---------------------------------------------------------------------------
__global__ __launch_bounds__(256) void recon_kernel(
    const float* __restrict__ x, const float* __restrict__ mean,
    const float* __restrict__ outk, const int* __restrict__ know,
    float* __restrict__ ws)
{
    const size_t total  = (size_t)KK * DD;
    const size_t stride = (size_t)gridDim.x * 256;
    float sq = 0.f;
    for (size_t e = (size_t)blockIdx.x * 256 + threadIdx.x; e < total; e += stride) {
        int kk = (int)(e >> 8);
        int d  = (int)(e & 255);
        int row = know[kk];
        float diff = x[(size_t)row * DD + d] - mean[d] - outk[e];
        sq += diff * diff;
    }
    for (int off = 16; off > 0; off >>= 1)
        sq += __shfl_down(sq, off, 32);
    if ((threadIdx.x & 31) == 0) atomicAdd(&ws[3], sq);
}

__global__ void finalize_kernel(const float* __restrict__ theta,
                                const float* __restrict__ eta,
                                const float* __restrict__ ws,
                                float* __restrict__ out)
{
    double dir   = (double)ws[0];
    double sumsq = (double)ws[1];
    double colsq = (double)ws[2];
    double recon = (double)ws[3];
    const double n = (double)NN;
    double avg = (n / (n - 1.0)) * sumsq - (1.0 / (n - 1.0)) * colsq;
    out[0] = (float)(-(dir + (double)eta[0] * recon + (double)theta[0] * avg));
}

extern "C" void kernel_launch(void* const* d_in, const int* in_sizes, int n_in,
                              void* d_out, int out_size, void* d_ws, size_t ws_size,
                              hipStream_t stream) {
    const float* x      = (const float*)d_in[0];   // [N,D]
    const float* L      = (const float*)d_in[1];   // [N,N]
    const float* mean   = (const float*)d_in[2];   // [D]
    const float* outk   = (const float*)d_in[3];   // [K,D]
    const int*   know   = (const int*)d_in[4];     // [K] (jax default int32)
    const float* theta  = (const float*)d_in[5];   // scalar
    const float* eta    = (const float*)d_in[6];   // scalar
    float* out = (float*)d_out;
    float* ws  = (float*)d_ws;

    (void)in_sizes; (void)n_in; (void)out_size; (void)ws_size;

    zero_ws_kernel<<<1, 512, 0, stream>>>(ws);
    gram_dirichlet_kernel<<<dim3(NTILES, 2), 256, 0, stream>>>(x, L, mean, ws);
    colsum_kernel<<<NN / ROWS_PER_CS, 256, 0, stream>>>(x, mean, ws);
    recon_kernel<<<160, 256, 0, stream>>>(x, mean, outk, know, ws);
    colsq_kernel<<<1, 256, 0, stream>>>(mean, ws);
    finalize_kernel<<<1, 1, 0, stream>>>(theta, eta, ws, out);
}